// ContourIntegrationLayer_46772193854045
// MI455X (gfx1250) — compile-verified
//
#include <hip/hip_runtime.h>
#include <cstdint>

// Problem constants (from reference): x(32,56,56,256) NHWC, kernel(3,3,256),
// depthwise 3x3 with center tap zeroed, SAME padding, + residual.
#define BATCH    32
#define H        56
#define W        56
#define CHN      256
#define TILE_W   14               // 56 = 4 tiles of 14
#define HALO_W   (TILE_W + 2)     // 16 cols staged per row
#define ROW_F    (HALO_W * CHN)   // 4096 floats per LDS row
#define CHUNKS   (ROW_F / 4)      // 1024 x b128 chunks per row
#define NTHREADS 256              // 8 waves of 32

__global__ __launch_bounds__(NTHREADS)
void dwconv3x3_masked_residual(const float* __restrict__ x,
                               const float* __restrict__ kern,
                               float* __restrict__ out) {
    __shared__ float smem[3 * ROW_F];   // 48 KB: 3 rows x 16 cols x 256 ch

    const int bid   = blockIdx.x;
    const int tile  = bid & 3;                 // 4 column tiles per row
    const int r     = (bid >> 2) % H;
    const int b     = bid / (4 * H);
    const int tile0 = tile * TILE_W;
    const int tid   = threadIdx.x;

    const int r0 = (r > 0)     ? r - 1 : 0;        // clamped (weights masked)
    const int r2 = (r < H - 1) ? r + 1 : H - 1;

    const float* base0 = x + ((size_t)(b * H + r0) * W) * CHN;
    const float* base1 = x + ((size_t)(b * H + r ) * W) * CHN;
    const float* base2 = x + ((size_t)(b * H + r2) * W) * CHN;

    // ---- Stage 3 input rows (with 1-col halo) into LDS via CDNA5 async
    // global->LDS B128 copies (ASYNCcnt-tracked, bypasses VGPR staging).
    #pragma unroll
    for (int row = 0; row < 3; ++row) {
        const float* base = (row == 0) ? base0 : (row == 1) ? base1 : base2;
        #pragma unroll
        for (int it = 0; it < CHUNKS / NTHREADS; ++it) {
            const int idx = it * NTHREADS + tid;    // 0..1023
            const int lc  = idx >> 6;               // local col 0..15
            const int sub = idx & 63;               // 16B chunk within column
            const int g   = tile0 + lc - 1;         // global image column
            float* lptr = &smem[row * ROW_F + idx * 4];
            const uint32_t lds_addr = (uint32_t)(uintptr_t)lptr; // LDS byte addr
            if (g >= 0 && g < W) {
                const uint32_t voff = (uint32_t)(g * (CHN * 4) + sub * 16);
                asm volatile("global_load_async_to_lds_b128 %0, %1, %2"
                             :: "v"(lds_addr), "v"(voff), "s"(base)
                             : "memory");
            } else {
                // SAME zero-padding for out-of-image halo columns
                *reinterpret_cast<float4*>(lptr) = make_float4(0.f, 0.f, 0.f, 0.f);
            }
        }
    }
    asm volatile("s_wait_asynccnt 0" ::: "memory");
    __syncthreads();

    // ---- Per-thread channel; row-boundary handling folded into weights.
    const int   ch  = tid;
    const float top = (r > 0)     ? 1.f : 0.f;
    const float bot = (r < H - 1) ? 1.f : 0.f;

    const float w00 = kern[0 * CHN + ch] * top;
    const float w01 = kern[1 * CHN + ch] * top;
    const float w02 = kern[2 * CHN + ch] * top;
    const float w10 = kern[3 * CHN + ch];
    // center tap (1,1) is masked to zero by the reference -> skipped
    const float w12 = kern[5 * CHN + ch];
    const float w20 = kern[6 * CHN + ch] * bot;
    const float w21 = kern[7 * CHN + ch] * bot;
    const float w22 = kern[8 * CHN + ch] * bot;

    const float* L0 = &smem[0 * ROW_F + ch];
    const float* L1 = &smem[1 * ROW_F + ch];
    const float* L2 = &smem[2 * ROW_F + ch];

    // Rolling 3-wide window: 3 LDS loads per output element.
    float aL = L0[0], aC = L0[CHN];
    float bL = L1[0], bC = L1[CHN];
    float cL = L2[0], cC = L2[CHN];

    float* op = out + (((size_t)(b * H + r) * W + tile0) * CHN) + ch;

    #pragma unroll
    for (int oc = 0; oc < TILE_W; ++oc) {
        const float aR = L0[(oc + 2) * CHN];
        const float bR = L1[(oc + 2) * CHN];
        const float cR = L2[(oc + 2) * CHN];

        float acc = bC;                    // residual (= x at center)
        acc = fmaf(w00, aL, acc);
        acc = fmaf(w01, aC, acc);
        acc = fmaf(w02, aR, acc);
        acc = fmaf(w10, bL, acc);
        acc = fmaf(w12, bR, acc);
        acc = fmaf(w20, cL, acc);
        acc = fmaf(w21, cC, acc);
        acc = fmaf(w22, cR, acc);

        // Non-temporal store: output is written once and never re-read, so
        // keep it out of L2 (192 MB) so the 103 MB input stays resident and
        // the 3x cross-block row reuse never goes back to HBM.
        __builtin_nontemporal_store(acc, &op[oc * CHN]);

        aL = aC; aC = aR;
        bL = bC; bC = bR;
        cL = cC; cC = cR;
    }
}

extern "C" void kernel_launch(void* const* d_in, const int* in_sizes, int n_in,
                              void* d_out, int out_size, void* d_ws, size_t ws_size,
                              hipStream_t stream) {
    const float* x    = (const float*)d_in[0];   // (32,56,56,256) f32
    const float* kern = (const float*)d_in[1];   // (3,3,256) f32
    float*       out  = (float*)d_out;           // (32,56,56,256) f32

    dim3 grid(BATCH * H * 4);   // one block per (batch, row, 14-col tile)
    dim3 block(NTHREADS);
    dwconv3x3_masked_residual<<<grid, block, 0, stream>>>(x, kern, out);
}